// SurfNet_45646912422004
// MI455X (gfx1250) — compile-verified
//
#include <hip/hip_runtime.h>

typedef __attribute__((ext_vector_type(2))) float v2f;
typedef __attribute__((ext_vector_type(8))) float v8f;

// ---------------------------------------------------------------------------
// Zero-fill (graph-capture-safe replacement for hipMemsetAsync)
// ---------------------------------------------------------------------------
__global__ void zero_f32(float* __restrict__ p, long n) {
    long i = (long)blockIdx.x * blockDim.x + threadIdx.x;
    long stride = (long)gridDim.x * blockDim.x;
    for (; i < n; i += stride) p[i] = 0.0f;
}

// ---------------------------------------------------------------------------
// Zero-pad W1 [3 x dOut] -> W1p [4 x dOut] (row 3 = zeros) so the layer-1
// GEMM runs with DIN=4 and no lane-varying predicates in the K-loop.
// ---------------------------------------------------------------------------
__global__ void pad_w1_kernel(const float* __restrict__ W1, float* __restrict__ W1p,
                              int dOut) {
    int i = blockIdx.x * blockDim.x + threadIdx.x;
    if (i >= 4 * dOut) return;
    int k = i / dOut;
    W1p[i] = (k < 3) ? W1[i] : 0.0f;
}

// ---------------------------------------------------------------------------
// Degree accumulation: deg_out[src[e]] += 1 ; deg_in[dst[e]] += 1
// ---------------------------------------------------------------------------
__global__ void degree_kernel(const int* __restrict__ src, const int* __restrict__ dst,
                              float* __restrict__ deg_out, float* __restrict__ deg_in,
                              int nE) {
    int e = blockIdx.x * blockDim.x + threadIdx.x;
    if (e >= nE) return;
    atomicAdd(&deg_out[src[e]], 1.0f);
    atomicAdd(&deg_in[dst[e]], 1.0f);
}

// norm = deg > 0 ? rsqrt(deg) : 0   (DGL norm='both')
__global__ void norm_kernel(const float* __restrict__ deg, float* __restrict__ norm, int n) {
    int i = blockIdx.x * blockDim.x + threadIdx.x;
    if (i >= n) return;
    float d = deg[i];
    norm[i] = (d > 0.0f) ? __frsqrt_rn(d) : 0.0f;
}

// ---------------------------------------------------------------------------
// Fused scatter: agg[dst[e], f] += (xa + xb + xc)[src[e], f] * norm_out[src[e]]
// One thread per (edge, 4-feature chunk). float4 loads when dIn % 4 == 0.
// aggStride lets layer 1 (dIn=3) write into a stride-4 zero-padded buffer.
// ---------------------------------------------------------------------------
__global__ void scatter_kernel(const float* __restrict__ xa, const float* __restrict__ xb,
                               const float* __restrict__ xc,
                               const int* __restrict__ src, const int* __restrict__ dst,
                               const float* __restrict__ norm_out, float* __restrict__ agg,
                               int nE, int dIn, int aggStride) {
    int nchunk = (dIn + 3) >> 2;
    long total = (long)nE * nchunk;
    long idx = (long)blockIdx.x * blockDim.x + threadIdx.x;
    if (idx >= total) return;
    int e = (int)(idx / nchunk);
    int c = (int)(idx % nchunk);
    int s = src[e];
    int d = dst[e];
    float w = norm_out[s];
    int f0 = c << 2;
    long sbase = (long)s * dIn;
    long dbase = (long)d * aggStride;
    if (f0 + 4 <= dIn) {
        // aligned vector path (dIn multiple of 4 for all layers except layer 1)
        float4 v = *(const float4*)(xa + sbase + f0);
        if (xb) {
            float4 u = *(const float4*)(xb + sbase + f0);
            v.x += u.x; v.y += u.y; v.z += u.z; v.w += u.w;
        }
        if (xc) {
            float4 u = *(const float4*)(xc + sbase + f0);
            v.x += u.x; v.y += u.y; v.z += u.z; v.w += u.w;
        }
        atomicAdd(agg + dbase + f0 + 0, v.x * w);
        atomicAdd(agg + dbase + f0 + 1, v.y * w);
        atomicAdd(agg + dbase + f0 + 2, v.z * w);
        atomicAdd(agg + dbase + f0 + 3, v.w * w);
    } else {
        // tail path (layer 1, dIn == 3)
        for (int f = f0; f < dIn; ++f) {
            float v = xa[sbase + f];
            if (xb) v += xb[sbase + f];
            if (xc) v += xc[sbase + f];
            atomicAdd(agg + dbase + f, v * w);
        }
    }
}

// ---------------------------------------------------------------------------
// WMMA GEMM: out = relu( norm_in[:,None] * (agg @ W) + bias )
//
// Each wave32 computes TWO 16x16 output tiles (a 32-row x 16-col super-tile)
// via V_WMMA_F32_16X16X4_F32, sharing the B fragment between both WMMAs:
// per K-step = 2x A b64 loads + 2x B b32 loads -> 2 WMMAs (load:wmma = 2).
//
// ISA VGPR layouts (CDNA5 ISA §7.12.2):
//   A (16x4, MxK):  lanes 0-15 -> M=lane,    a[j] = A[M][K=j]       (K=0,1)
//                   lanes 16-31-> M=lane-16, a[j] = A[M][K=2+j]     (K=2,3)
//   B (4x16, KxN):  mirrored: lanes 0-15 hold K=0,1; lanes 16-31 hold K=2,3,
//                   N = lane & 15 within one VGPR pair
//   C/D (16x16):    VGPR r: lanes 0-15 -> M=r, lanes 16-31 -> M=r+8; N=lane&15
//
// The per-row norm_in scale is hoisted out of the K-loop
// (diag(n)*A*W == diag(n)*(A*W)) into the epilogue, fetched as b128 vectors.
// Epilogue bounds check is wave-uniform: fast path (always taken, 20000%32==0)
// has no EXEC churn — stores at immediate offsets off two base pointers.
// ---------------------------------------------------------------------------
template <int DIN, int DOUT>
__global__ __launch_bounds__(128) void gemm_bias_relu_wmma(
        const float* __restrict__ agg, const float* __restrict__ norm_in,
        const float* __restrict__ W, const float* __restrict__ bias,
        float* __restrict__ out, int nNodes) {
    const int mt   = blockIdx.x;                             // 32-row super-tile
    const int nt   = blockIdx.y * blockDim.y + threadIdx.y;  // N tile (16 cols)
    const int lane = threadIdx.x;                            // 0..31 (wave32)
    const int half = lane >> 4;                              // 0: K=0,1 | 1: K=2,3
    const int l    = lane & 15;

    int row0 = mt * 32 + l;                                  // A row, tile 0
    int row1 = row0 + 16;                                    // A row, tile 1
    if (row0 >= nNodes) row0 = nNodes - 1;                   // clamp (stores guarded)
    if (row1 >= nNodes) row1 = nNodes - 1;
    const int col = nt * 16 + l;                             // B/D column

    const float* __restrict__ arow0 = agg + (long)row0 * DIN;
    const float* __restrict__ arow1 = agg + (long)row1 * DIN;
    const float* __restrict__ wcol  = W + col;

    v8f acc0 = {};
    v8f acc1 = {};
#pragma unroll 8
    for (int k0 = 0; k0 < DIN; k0 += 4) {
        const int ka = k0 + half * 2;                        // this half-wave's K base
        v2f a0 = *(const v2f*)(arow0 + ka);                  // one b64 load each
        v2f a1 = *(const v2f*)(arow1 + ka);
        v2f b;
        b.x = wcol[(long)ka * DOUT];
        b.y = wcol[(long)ka * DOUT + DOUT];
        acc0 = __builtin_amdgcn_wmma_f32_16x16x4_f32(
            false, a0, false, b, (short)0, acc0, false, false);
        acc1 = __builtin_amdgcn_wmma_f32_16x16x4_f32(
            false, a1, false, b, (short)0, acc1, false, false);
    }

    const float bb    = bias[col];
    const int   mbase = mt * 32 + half * 8;                  // D rows: mbase+r (+16 tile1)

    if (mt * 32 + 32 <= nNodes) {
        // -------- fast path (wave-uniform, always taken for N % 32 == 0) ----
        const float4 na_lo = *(const float4*)(norm_in + mbase);
        const float4 na_hi = *(const float4*)(norm_in + mbase + 4);
        const float4 nb_lo = *(const float4*)(norm_in + mbase + 16);
        const float4 nb_hi = *(const float4*)(norm_in + mbase + 20);
        const float nv0[8] = {na_lo.x, na_lo.y, na_lo.z, na_lo.w,
                              na_hi.x, na_hi.y, na_hi.z, na_hi.w};
        const float nv1[8] = {nb_lo.x, nb_lo.y, nb_lo.z, nb_lo.w,
                              nb_hi.x, nb_hi.y, nb_hi.z, nb_hi.w};
        float* __restrict__ o0 = out + (long)mbase * DOUT + col;
        float* __restrict__ o1 = o0 + 16L * DOUT;
#pragma unroll
        for (int r = 0; r < 8; ++r) {                        // immediate-offset stores
            float v0 = fmaf(acc0[r], nv0[r], bb);
            float v1 = fmaf(acc1[r], nv1[r], bb);
            o0[r * DOUT] = v0 > 0.0f ? v0 : 0.0f;
            o1[r * DOUT] = v1 > 0.0f ? v1 : 0.0f;
        }
    } else {
        // -------- guarded tail (robustness only) ---------------------------
#pragma unroll
        for (int r = 0; r < 8; ++r) {
            const int m0 = mbase + r;
            const int m1 = m0 + 16;
            if (m0 < nNodes) {
                float v = fmaf(acc0[r], norm_in[m0], bb);
                out[(long)m0 * DOUT + col] = v > 0.0f ? v : 0.0f;
            }
            if (m1 < nNodes) {
                float v = fmaf(acc1[r], norm_in[m1], bb);
                out[(long)m1 * DOUT + col] = v > 0.0f ? v : 0.0f;
            }
        }
    }
}

// ---------------------------------------------------------------------------
// Host-side orchestration (graph-capture safe: only kernel launches on stream)
// ---------------------------------------------------------------------------
static inline void launch_zero(float* p, long n, hipStream_t s) {
    int blocks = (int)((n + 255) / 256);
    if (blocks > 4096) blocks = 4096;
    zero_f32<<<blocks, 256, 0, s>>>(p, n);
}

template <int DIN, int DOUT>
static void run_layer(const float* xa, const float* xb, const float* xc,
                      const int* src, const int* dst,
                      const float* norm_out, const float* norm_in,
                      const float* W, const float* bias,
                      float* agg, float* out,
                      int nNodes, int nE, int dInRead, hipStream_t s) {
    launch_zero(agg, (long)nNodes * DIN, s);

    int nchunk = (dInRead + 3) >> 2;
    long total = (long)nE * nchunk;
    scatter_kernel<<<(int)((total + 255) / 256), 256, 0, s>>>(
        xa, xb, xc, src, dst, norm_out, agg, nE, dInRead, DIN);

    // 4 waves/block, each wave owns a 32x16 output super-tile
    dim3 grid((nNodes + 31) / 32, DOUT / 64);
    dim3 block(32, 4);
    gemm_bias_relu_wmma<DIN, DOUT><<<grid, block, 0, s>>>(agg, norm_in, W, bias,
                                                          out, nNodes);
}

extern "C" void kernel_launch(void* const* d_in, const int* in_sizes, int n_in,
                              void* d_out, int out_size, void* d_ws, size_t ws_size,
                              hipStream_t stream) {
    const float* features = (const float*)d_in[0];     // [N, 3]
    const int*   src      = (const int*)d_in[1];       // [E]
    const int*   dst      = (const int*)d_in[2];       // [E]
    const float* W[6];
    const float* B[6];
    for (int i = 0; i < 6; ++i) {
        W[i] = (const float*)d_in[3 + 2 * i];
        B[i] = (const float*)d_in[4 + 2 * i];
    }
    const int N = in_sizes[0] / 3;                     // 20000
    const int E = in_sizes[1];                         // 320000

    // ---- workspace carve-up (floats) ----
    float* ws = (float*)d_ws;
    float* deg_out  = ws; ws += N;                     // contiguous with deg_in
    float* deg_in   = ws; ws += N;
    float* norm_out = ws; ws += N;
    float* norm_in  = ws; ws += N;
    float* w1pad    = ws; ws += 4 * 64;                // zero-padded W1 [4 x 64]
    float* agg      = ws; ws += (long)N * 256;
    float* x1       = ws; ws += (long)N * 64;
    float* x2       = ws; ws += (long)N * 128;
    float* x3       = ws; ws += (long)N * 256;
    float* x4       = ws; ws += (long)N * 256;
    float* x5       = ws; ws += (long)N * 256;
    float* x6       = (float*)d_out;                   // [N, 256]

    // ---- degree norms + W1 padding ----
    launch_zero(deg_out, 2L * N, stream);              // deg_out and deg_in adjacent
    degree_kernel<<<(E + 255) / 256, 256, 0, stream>>>(src, dst, deg_out, deg_in, E);
    norm_kernel<<<(N + 255) / 256, 256, 0, stream>>>(deg_out, norm_out, N);
    norm_kernel<<<(N + 255) / 256, 256, 0, stream>>>(deg_in, norm_in, N);
    pad_w1_kernel<<<1, 256, 0, stream>>>(W[0], w1pad, 64);

    // ---- 6 graph-conv layers (layers 5/6 sum residuals in the scatter) ----
    run_layer<4, 64>(features, nullptr, nullptr, src, dst, norm_out, norm_in,
                     w1pad, B[0], agg, x1, N, E, /*dInRead=*/3, stream);
    run_layer<64, 128>(x1, nullptr, nullptr, src, dst, norm_out, norm_in,
                       W[1], B[1], agg, x2, N, E, 64, stream);
    run_layer<128, 256>(x2, nullptr, nullptr, src, dst, norm_out, norm_in,
                        W[2], B[2], agg, x3, N, E, 128, stream);
    run_layer<256, 256>(x3, nullptr, nullptr, src, dst, norm_out, norm_in,
                        W[3], B[3], agg, x4, N, E, 256, stream);
    run_layer<256, 256>(x4, x3, nullptr, src, dst, norm_out, norm_in,
                        W[4], B[4], agg, x5, N, E, 256, stream);
    run_layer<256, 256>(x3, x4, x5, src, dst, norm_out, norm_in,
                        W[5], B[5], agg, x6, N, E, 256, stream);
}